// Multi_Modal_17858474017317
// MI455X (gfx1250) — compile-verified
//
#include <hip/hip_runtime.h>
#include <math.h>

typedef float v2f __attribute__((ext_vector_type(2)));
typedef float v8f __attribute__((ext_vector_type(8)));

#define N_NODES  100000
#define N_EDGES  1600000
#define N_GRAPHS 1024
#define HID      128

// ---------------------------------------------------------------- utilities
__global__ void k_fill(float* __restrict__ p, float v, int n) {
  int i = blockIdx.x * blockDim.x + threadIdx.x;
  if (i < n) p[i] = v;
}

__global__ void k_deg_edges(const int* __restrict__ ei, float* __restrict__ deg) {
  int e = blockIdx.x * blockDim.x + threadIdx.x;
  if (e < N_EDGES) atomicAdd(&deg[ei[N_EDGES + e]], 1.0f);
}

__global__ void k_rsqrt_inplace(float* __restrict__ p, int n) {
  int i = blockIdx.x * blockDim.x + threadIdx.x;
  if (i < n) p[i] = rsqrtf(p[i]);
}

// ------------------------------------------------- fp32 WMMA GEMM: C = A @ W
// Compile-time K,N so the unrolled inner loop uses immediate-offset global
// loads for B and constant-offset DS loads for A (no 64-bit MAD chains).
// A: (M x K) dense row-major. W: (K x N) row-major.
// C written at C[row*ldc + coff + col]. One block = 16 rows x 128 cols,
// 8 waves, wave w owns 16 columns. grid = (M/16, N/128).
template <int K, int N, bool BIAS, bool RELU>
__global__ __launch_bounds__(256)
void wmma_gemm_f32(const float* __restrict__ A, const float* __restrict__ W,
                   const float* __restrict__ bias, float* __restrict__ C,
                   int ldc, int coff)
{
  constexpr int Kp = K + 4;               // padded stride: no LDS bank conflicts
  __shared__ float As[16 * Kp];

  // cooperative stage of the 16xK A tile into LDS (float4, 16 threads/row)
  {
    const float* Ablk = A + (size_t)blockIdx.x * 16 * K;
    int r = threadIdx.x >> 4;                       // 0..15
    const float4* src = (const float4*)(Ablk + (size_t)r * K);
    float4*       dst = (float4*)(&As[r * Kp]);
#pragma unroll
    for (int j = (threadIdx.x & 15); j < (K >> 2); j += 16) dst[j] = src[j];
  }
  __syncthreads();

  int wv   = threadIdx.x >> 5;
  int col0 = blockIdx.y * 128 + wv * 16;            // always < N (N % 128 == 0)

  int lane = threadIdx.x & 31;
  int half = lane >> 4;          // 0: K pair {0,1}, 1: K pair {2,3}
  int r    = lane & 15;          // A row (SRC0) / B column (SRC1)

  const float* Ar = &As[r * Kp + 2 * half];
  const float* Wp = W + (size_t)(2 * half) * N + col0 + r;  // immediate offsets below

  v8f acc = {};
#pragma unroll
  for (int k = 0; k < K; k += 4) {
    v2f a, b;
    a[0] = Ar[k];
    a[1] = Ar[k + 1];
    b[0] = Wp[(size_t)k * N];
    b[1] = Wp[(size_t)(k + 1) * N];
    acc = __builtin_amdgcn_wmma_f32_16x16x4_f32(false, a, false, b,
                                                (short)0, acc, false, false);
  }

  // C/D layout: VGPR i, lane -> row = i + 8*half, col = lane&15
  float bv = 0.0f;
  if (BIAS) bv = bias[col0 + r];
  float* Cp = C + (size_t)(blockIdx.x * 16 + 8 * half) * ldc + coff + col0 + r;
#pragma unroll
  for (int i = 0; i < 8; ++i) {
    float v = acc[i] + bv;
    if (RELU) v = fmaxf(v, 0.0f);
    Cp[(size_t)i * ldc] = v;
  }
}

// ------------------------------------------------- edge scatter (wave/edge)
__global__ __launch_bounds__(256)
void k_scatter(const int* __restrict__ ei, const float* __restrict__ dinv,
               const float* __restrict__ HW, float* __restrict__ agg)
{
  int e = blockIdx.x * 8 + (threadIdx.x >> 5);
  if (e >= N_EDGES) return;
  int lane = threadIdx.x & 31;
  int s = ei[e];
  int d = ei[N_EDGES + e];
  float nrm = dinv[s] * dinv[d];
  float4 v = ((const float4*)(HW + (size_t)s * HID))[lane];
  float* a = agg + (size_t)d * HID + lane * 4;
  atomicAdd(a + 0, nrm * v.x);
  atomicAdd(a + 1, nrm * v.y);
  atomicAdd(a + 2, nrm * v.z);
  atomicAdd(a + 3, nrm * v.w);
}

// H[i] = relu(agg[i] + 2*dinv[i]^2*HW[i] + b)   (two self-loops folded in)
__global__ __launch_bounds__(256)
void k_finalize(const float* __restrict__ agg, const float* __restrict__ HW,
                const float* __restrict__ dinv, const float* __restrict__ bias,
                float* __restrict__ Hout)
{
  int i = blockIdx.x * 8 + (threadIdx.x >> 5);
  if (i >= N_NODES) return;
  int lane = threadIdx.x & 31;
  float di = dinv[i];
  float sl = 2.0f * di * di;
  float4 av = ((const float4*)(agg + (size_t)i * HID))[lane];
  float4 hv = ((const float4*)(HW + (size_t)i * HID))[lane];
  float4 bv = ((const float4*)bias)[lane];
  float4 o;
  o.x = fmaxf(av.x + sl * hv.x + bv.x, 0.0f);
  o.y = fmaxf(av.y + sl * hv.y + bv.y, 0.0f);
  o.z = fmaxf(av.z + sl * hv.z + bv.z, 0.0f);
  o.w = fmaxf(av.w + sl * hv.w + bv.w, 0.0f);
  ((float4*)(Hout + (size_t)i * HID))[lane] = o;
}

// ------------------------------------------------- global mean/max pooling
__global__ __launch_bounds__(256)
void k_pool(const float* __restrict__ H, const int* __restrict__ batch,
            float* __restrict__ sum, int* __restrict__ mx, float* __restrict__ cnt)
{
  int i = blockIdx.x * 8 + (threadIdx.x >> 5);
  if (i >= N_NODES) return;
  int lane = threadIdx.x & 31;
  int g = batch[i];
  float4 v = ((const float4*)(H + (size_t)i * HID))[lane];
  float* s = sum + (size_t)g * HID + lane * 4;
  atomicAdd(s + 0, v.x);
  atomicAdd(s + 1, v.y);
  atomicAdd(s + 2, v.z);
  atomicAdd(s + 3, v.w);
  // post-ReLU values are >= 0 so float bit pattern order == signed int order
  int* m = mx + (size_t)g * HID + lane * 4;
  atomicMax(m + 0, __float_as_int(v.x));
  atomicMax(m + 1, __float_as_int(v.y));
  atomicMax(m + 2, __float_as_int(v.z));
  atomicMax(m + 3, __float_as_int(v.w));
  if (lane == 0) atomicAdd(&cnt[g], 1.0f);
}

// xcat[:, 0:128] = mean, xcat[:, 128:256] = max  (cols 256:384 come from tab2)
__global__ void k_assemble(const float* __restrict__ sum, const int* __restrict__ mx,
                           const float* __restrict__ cnt, float* __restrict__ xcat)
{
  int idx = blockIdx.x * blockDim.x + threadIdx.x;
  if (idx >= N_GRAPHS * HID) return;
  int g = idx >> 7, f = idx & 127;
  float c = fmaxf(cnt[g], 1.0f);
  xcat[(size_t)g * 384 + f]       = sum[idx] / c;
  xcat[(size_t)g * 384 + 128 + f] = __int_as_float(mx[idx]);
}

// out[g] = sigmoid(h2[g] . Wf3 + bf3)
__global__ void k_head_final(const float* __restrict__ h2, const float* __restrict__ Wf3,
                             const float* __restrict__ bf3, float* __restrict__ out)
{
  int g = blockIdx.x * blockDim.x + threadIdx.x;
  if (g >= N_GRAPHS) return;
  float acc = bf3[0];
  const float* hp = h2 + (size_t)g * 128;
  for (int f = 0; f < 128; ++f) acc += hp[f] * Wf3[f];
  out[g] = 1.0f / (1.0f + expf(-acc));
}

// ---------------------------------------------------------------- launcher
extern "C" void kernel_launch(void* const* d_in, const int* in_sizes, int n_in,
                              void* d_out, int out_size, void* d_ws, size_t ws_size,
                              hipStream_t stream)
{
  (void)in_sizes; (void)n_in; (void)out_size; (void)ws_size;
  const float* x_graph = (const float*)d_in[0];
  const int*   ei      = (const int*)d_in[1];
  const int*   batch   = (const int*)d_in[2];
  const float* x_tab   = (const float*)d_in[3];
  const float* Wg[4] = {(const float*)d_in[4], (const float*)d_in[6],
                        (const float*)d_in[8], (const float*)d_in[10]};
  const float* bg[4] = {(const float*)d_in[5], (const float*)d_in[7],
                        (const float*)d_in[9], (const float*)d_in[11]};
  const float* Wt1 = (const float*)d_in[12], *bt1 = (const float*)d_in[13];
  const float* Wt2 = (const float*)d_in[14], *bt2 = (const float*)d_in[15];
  const float* Wf1 = (const float*)d_in[16], *bf1 = (const float*)d_in[17];
  const float* Wf2 = (const float*)d_in[18], *bf2 = (const float*)d_in[19];
  const float* Wf3 = (const float*)d_in[20], *bf3 = (const float*)d_in[21];
  float* out = (float*)d_out;

  char* ws = (char*)d_ws;
  size_t off = 0;
  auto alloc = [&](size_t bytes) -> void* {
    off = (off + 255) & ~(size_t)255;
    void* p = (void*)(ws + off);
    off += bytes;
    return p;
  };
  float* dinv = (float*)alloc((size_t)N_NODES * 4);                  // deg -> dinv in place
  float* HW   = (float*)alloc((size_t)N_NODES * HID * 4);
  float* AGG  = (float*)alloc((size_t)N_NODES * HID * 4);
  float* Hb   = (float*)alloc((size_t)N_NODES * HID * 4);
  float* sum  = (float*)alloc((size_t)N_GRAPHS * HID * 4);
  int*   mxb  = (int*)  alloc((size_t)N_GRAPHS * HID * 4);
  float* cnt  = (float*)alloc((size_t)N_GRAPHS * 4);
  float* t1   = (float*)alloc((size_t)N_GRAPHS * 256 * 4);
  float* xcat = (float*)alloc((size_t)N_GRAPHS * 384 * 4);
  float* hh1  = (float*)alloc((size_t)N_GRAPHS * 256 * 4);
  float* hh2  = (float*)alloc((size_t)N_GRAPHS * 128 * 4);

  // --- symmetric normalization: deg = 2 (two self-loops) + in-edges
  k_fill<<<(N_NODES + 255) / 256, 256, 0, stream>>>(dinv, 2.0f, N_NODES);
  k_deg_edges<<<(N_EDGES + 255) / 256, 256, 0, stream>>>(ei, dinv);
  k_rsqrt_inplace<<<(N_NODES + 255) / 256, 256, 0, stream>>>(dinv, N_NODES);

  // --- 4 GCN layers: GEMM (WMMA) -> zero -> scatter -> finalize(+self loops, bias, relu)
  for (int l = 0; l < 4; ++l) {
    if (l == 0) {
      wmma_gemm_f32<64, 128, false, false><<<dim3(N_NODES / 16, 1), 256, 0, stream>>>(
          x_graph, Wg[0], nullptr, HW, 128, 0);
    } else {
      wmma_gemm_f32<128, 128, false, false><<<dim3(N_NODES / 16, 1), 256, 0, stream>>>(
          Hb, Wg[l], nullptr, HW, 128, 0);
    }
    hipMemsetAsync(AGG, 0, (size_t)N_NODES * HID * 4, stream);
    k_scatter<<<(N_EDGES + 7) / 8, 256, 0, stream>>>(ei, dinv, HW, AGG);
    k_finalize<<<(N_NODES + 7) / 8, 256, 0, stream>>>(AGG, HW, dinv, bg[l], Hb);
  }

  // --- global mean/max pooling
  hipMemsetAsync(sum, 0, (size_t)N_GRAPHS * HID * 4, stream);
  hipMemsetAsync(mxb, 0, (size_t)N_GRAPHS * HID * 4, stream);
  hipMemsetAsync(cnt, 0, (size_t)N_GRAPHS * 4, stream);
  k_pool<<<(N_NODES + 7) / 8, 256, 0, stream>>>(Hb, batch, sum, mxb, cnt);
  k_assemble<<<(N_GRAPHS * HID) / 256, 256, 0, stream>>>(sum, mxb, cnt, xcat);

  // --- tabular branch: relu(x_tab@Wt1+bt1) -> relu(@Wt2+bt2) into xcat[:,256:384]
  wmma_gemm_f32<200, 256, true, true><<<dim3(N_GRAPHS / 16, 2), 256, 0, stream>>>(
      x_tab, Wt1, bt1, t1, 256, 0);
  wmma_gemm_f32<256, 128, true, true><<<dim3(N_GRAPHS / 16, 1), 256, 0, stream>>>(
      t1, Wt2, bt2, xcat, 384, 256);

  // --- fused head
  wmma_gemm_f32<384, 256, true, true><<<dim3(N_GRAPHS / 16, 2), 256, 0, stream>>>(
      xcat, Wf1, bf1, hh1, 256, 0);
  wmma_gemm_f32<256, 128, true, true><<<dim3(N_GRAPHS / 16, 1), 256, 0, stream>>>(
      hh1, Wf2, bf2, hh2, 128, 0);
  k_head_final<<<(N_GRAPHS + 255) / 256, 256, 0, stream>>>(hh2, Wf3, bf3, out);
}